// RegionSelector_72533407695358
// MI455X (gfx1250) — compile-verified
//
#include <hip/hip_runtime.h>

typedef __attribute__((ext_vector_type(2))) float v2f;
typedef __attribute__((ext_vector_type(4))) float v4f;
typedef __attribute__((ext_vector_type(8))) float v8f;

#define GRIDN 4
#define WIN   3
#define NW    2          // GRIDN - WIN + 1
#define IMH   512
#define IMW   512
#define GH    128        // IMH / GRIDN
#define GW    128        // IMW / GRIDN

// ---------------------------------------------------------------------------
// Kernel 1: one 256-thread block per (batch, grid-cell).
// Streams a 128x128 f32 tile with NT b128 loads (constant 16KB stride ->
// immediate-offset clause loads) and accumulates the tile sum on the matrix
// pipe: D = A x ones + C  (V_WMMA_F32_16X16X4_F32).
// ---------------------------------------------------------------------------
__global__ __launch_bounds__(256) void RegionSelector_cellsum_kernel(
    const float* __restrict__ in, float* __restrict__ cell_sums) {
  const int bid  = blockIdx.x;
  const int b    = bid >> 4;        // batch
  const int cell = bid & 15;        // gy*4 + gx
  const int gy   = cell >> 2;
  const int gx   = cell & 3;

  const int tid  = threadIdx.x;
  const int wave = tid >> 5;        // 0..7
  const int lane = tid & 31;        // 0..31

  // Per-thread base: wave w starts at row w, lane covers 16B of the 512B row
  // segment. Each iteration advances exactly 8 rows = 8*IMW floats = 16 KB,
  // which the compiler folds into the 24-bit immediate offset of each load.
  const float* p = in + (size_t)b * (IMH * IMW) + (size_t)(gy * GH + wave) * IMW
                      + gx * GW + lane * 4;

  v8f acc = {};                      // running C/D accumulator (16x16 f32)
  v2f ones; ones.x = 1.0f; ones.y = 1.0f;   // all-ones B (4x16) -> layout-agnostic

#pragma unroll
  for (int it = 0; it < 16; ++it) {
    v4f v = __builtin_nontemporal_load((const v4f*)(p + (size_t)it * (8 * IMW)));
    v2f a0; a0.x = v.x; a0.y = v.y;
    v2f a1; a1.x = v.z; a1.y = v.w;
    // D[m][n] = sum_k A[m][k] + C[m][n]; with B==1 every column is identical.
    acc = __builtin_amdgcn_wmma_f32_16x16x4_f32(false, a0, false, ones,
                                                (short)0, acc, false, false);
    acc = __builtin_amdgcn_wmma_f32_16x16x4_f32(false, a1, false, ones,
                                                (short)0, acc, false, false);
  }

  // Per-lane column sum of D: lanes 0-15 hold M=0..7, lanes 16-31 hold M=8..15.
  float s = acc[0] + acc[1] + acc[2] + acc[3] + acc[4] + acc[5] + acc[6] + acc[7];
  // Add the opposite half-wave (wave32): every lane now holds the wave total.
  s += __shfl_xor(s, 16, 32);

  __shared__ float wsum[8];          // 8 waves per 256-thread block on wave32
  if (lane == 0) wsum[wave] = s;
  __syncthreads();
  if (tid == 0) {
    float t = 0.0f;
#pragma unroll
    for (int i = 0; i < 8; ++i) t += wsum[i];
    cell_sums[bid] = t;
  }
}

// ---------------------------------------------------------------------------
// Kernel 2: one thread per batch. 3x3 window sums over the 4x4 cell grid,
// argmax of the 4 windows (first-index tie-break == jax.lax.top_k), emit
// (row, col) as int32.
// ---------------------------------------------------------------------------
__global__ __launch_bounds__(128) void RegionSelector_select_kernel(
    const float* __restrict__ cell_sums, int* __restrict__ out, int nbatch) {
  const int b = blockIdx.x * blockDim.x + threadIdx.x;
  if (b >= nbatch) return;

  float g[16];
#pragma unroll
  for (int i = 0; i < 16; ++i) g[i] = cell_sums[b * 16 + i];

  float best  = -__builtin_inff();
  int   besti = 0;
#pragma unroll
  for (int r = 0; r < NW; ++r) {
#pragma unroll
    for (int c = 0; c < NW; ++c) {
      float w = 0.0f;
#pragma unroll
      for (int dr = 0; dr < WIN; ++dr)
#pragma unroll
        for (int dc = 0; dc < WIN; ++dc)
          w += g[(r + dr) * GRIDN + (c + dc)];
      const int idx = r * NW + c;
      if (w > best) { best = w; besti = idx; }   // strict '>' => lowest-index tie-break
    }
  }
  out[b * 2 + 0] = besti / NW;   // row
  out[b * 2 + 1] = besti % NW;   // col
}

extern "C" void kernel_launch(void* const* d_in, const int* in_sizes, int n_in,
                              void* d_out, int out_size, void* d_ws, size_t ws_size,
                              hipStream_t stream) {
  const float* in = (const float*)d_in[0];
  const int nbatch = in_sizes[0] / (IMH * IMW);   // 128 for the reference setup

  float* cell_sums = (float*)d_ws;                 // nbatch*16 floats scratch
  int*   out       = (int*)d_out;                  // [B, 1, 2] int32

  RegionSelector_cellsum_kernel<<<dim3(nbatch * 16), dim3(256), 0, stream>>>(in, cell_sums);

  const int threads = 128;
  const int blocks  = (nbatch + threads - 1) / threads;
  RegionSelector_select_kernel<<<dim3(blocks), dim3(threads), 0, stream>>>(cell_sums, out, nbatch);
}